// GCNSampling_30322469110460
// MI455X (gfx1250) — compile-verified
//
#include <hip/hip_runtime.h>
#include <hip/hip_bf16.h>
#include <stdint.h>

typedef __attribute__((ext_vector_type(2))) float v2f;
typedef __attribute__((ext_vector_type(8))) float v8f;

#define N0 200000
#define N1 50000
#define N2 10000
#define IN_FEATS 602
#define HIDDEN 128
#define CLASSES 41
#define FANOUT 25

// chunks 0..8 (kb = 0..512) are fully in-bounds: 9*64 = 576 <= 602
#define NASYNC_CHUNKS 9
#define TAIL_KB 576

// ---------------------------------------------------------------------------
// Kernel 1: h1[i] = relu(W1 @ features[map1[i]] + b1)   (gather-before-GEMM)
// 256 threads = 8 waves; block owns 16 rows of h1; wave w owns hidden cols
// [16w,16w+16). K staged in 64-wide chunks through DOUBLE-BUFFERED LDS via
// async global->LDS DMA (ASYNCcnt), overlapped with f32 WMMA 16x16x4 compute.
// ---------------------------------------------------------------------------
__global__ __launch_bounds__(256) void gcn_l1_gather_gemm_relu(
    const float* __restrict__ feat, const float* __restrict__ W1,
    const float* __restrict__ b1, const int* __restrict__ map1,
    float* __restrict__ h1)
{
    __shared__ float Alds[2][16][66];    // 16 rows x 64 k (pad->66), x2 buffers
    __shared__ float Blds[2][128][66];   // W1 chunk: 128 hidden x 64 k, x2
    __shared__ int   rows[16];

    const int tid   = threadIdx.x;
    const int lane  = tid & 31;
    const int wave  = tid >> 5;
    const int hi    = lane >> 4;         // half-wave select (ISA WMMA layout)
    const int lm    = lane & 15;
    const int mbase = blockIdx.x * 16;
    const int n0    = wave * 16;

    if (tid < 16) rows[tid] = map1[mbase + tid];
    __syncthreads();                     // rows[] needed for async addresses

    // issue one chunk's staging as async global->LDS DMA: 18 b64 ops/thread
    auto issue_async = [&](int kb, int b) {
        // A tile: 16x64 floats = 512 float2, 2 per thread
        #pragma unroll
        for (int i = 0; i < 2; ++i) {
            int e  = tid + 256 * i;
            int m  = e >> 5;
            int kk = (e & 31) * 2;
            unsigned lo   = (unsigned)(uintptr_t)&Alds[b][m][kk];
            const float* g = feat + (long)rows[m] * IN_FEATS + kb + kk;
            asm volatile("global_load_async_to_lds_b64 %0, %1, off"
                         :: "v"(lo), "v"(g) : "memory");
        }
        // W1 chunk: 128x64 floats = 4096 float2, 16 per thread
        #pragma unroll
        for (int i = 0; i < 16; ++i) {
            int e  = tid + 256 * i;
            int n  = e >> 5;
            int kk = (e & 31) * 2;
            unsigned lo   = (unsigned)(uintptr_t)&Blds[b][n][kk];
            const float* g = W1 + n * IN_FEATS + kb + kk;
            asm volatile("global_load_async_to_lds_b64 %0, %1, off"
                         :: "v"(lo), "v"(g) : "memory");
        }
    };

    v8f acc = {};

    auto compute_chunk = [&](int b) {
        #pragma unroll
        for (int ks = 0; ks < 16; ++ks) {
            int k0 = ks * 4 + 2 * hi;
            v2f a = *(const v2f*)&Alds[b][lm][k0];          // A[m=lm][k]
            v2f bb = *(const v2f*)&Blds[b][n0 + lm][k0];    // B[k][n]=W1[n][k]
            acc = __builtin_amdgcn_wmma_f32_16x16x4_f32(
                false, a, false, bb, (short)0, acc, false, false);
        }
    };

    // software pipeline: DMA chunk c+1 while WMMAs consume chunk c
    issue_async(0, 0);
    for (int c = 0; c < NASYNC_CHUNKS; ++c) {
        const int cur = c & 1;
        if (c + 1 < NASYNC_CHUNKS) {
            issue_async((c + 1) * 64, cur ^ 1);
            // 18 just-issued ops outstanding; in-order completion => chunk c done
            asm volatile("s_wait_asynccnt 18" ::: "memory");
        } else {
            asm volatile("s_wait_asynccnt 0" ::: "memory");
        }
        __syncthreads();
        compute_chunk(cur);
        __syncthreads();                 // all waves done before buffer reuse
    }

    // ragged tail chunk (kb=576, 26 valid cols): guarded VGPR path, zero-fill
    {
        const int b = 1;                 // chunk 8 used buf 0; buf 1 is free
        #pragma unroll
        for (int i = 0; i < 2; ++i) {
            int e  = tid + 256 * i;
            int m  = e >> 5;
            int kk = (e & 31) * 2;
            v2f val = {0.f, 0.f};
            if (TAIL_KB + kk + 2 <= IN_FEATS)
                val = *(const v2f*)(feat + (long)rows[m] * IN_FEATS + TAIL_KB + kk);
            *(v2f*)&Alds[b][m][kk] = val;
        }
        #pragma unroll
        for (int i = 0; i < 16; ++i) {
            int e  = tid + 256 * i;
            int n  = e >> 5;
            int kk = (e & 31) * 2;
            v2f val = {0.f, 0.f};
            if (TAIL_KB + kk + 2 <= IN_FEATS)
                val = *(const v2f*)(W1 + n * IN_FEATS + TAIL_KB + kk);
            *(v2f*)&Blds[b][n][kk] = val;
        }
        __syncthreads();
        compute_chunk(b);
    }

    // epilogue: bias + relu; D element acc[r] <-> (m = r + 8*hi, n = lm)
    float bias = b1[n0 + lm];
    #pragma unroll
    for (int r = 0; r < 8; ++r) {
        int m = r + 8 * hi;
        float v = acc[r] + bias;
        v = v > 0.f ? v : 0.f;
        h1[(long)(mbase + m) * HIDDEN + n0 + lm] = v;
    }
}

// ---------------------------------------------------------------------------
// Kernel 2: agg = mean_j h1[neigh_idx[n][j]] ; out = agg @ W2^T + b2
// Block = 16 seed nodes, 128 threads = 4 waves; waves 0..2 each do a
// 16-class WMMA tile over K=128 (classes padded 41->48 with zeros).
// ---------------------------------------------------------------------------
__global__ __launch_bounds__(128) void gcn_l2_agg_gemm(
    const float* __restrict__ h1, const float* __restrict__ W2,
    const float* __restrict__ b2, const int* __restrict__ neigh,
    float* __restrict__ out)
{
    __shared__ float Agg[16][132];   // 16 nodes x 128 h, pad->132
    __shared__ float W2s[48][132];   // zero-padded class rows 41..47
    __shared__ int   nidx[16 * FANOUT];

    const int tid   = threadIdx.x;
    const int lane  = tid & 31;
    const int wave  = tid >> 5;
    const int hi    = lane >> 4;
    const int lm    = lane & 15;
    const int mbase = blockIdx.x * 16;

    for (int e = tid; e < 16 * FANOUT; e += 128)
        nidx[e] = neigh[mbase * FANOUT + e];

    // stage W2 (41x128) zero-padded to 48 rows: 3072 float2, 24/thread
    #pragma unroll
    for (int i = 0; i < 24; ++i) {
        int e  = tid + 128 * i;
        int c  = e >> 6;                 // 64 float2 per row
        int kk = (e & 63) * 2;
        v2f val = {0.f, 0.f};
        if (c < CLASSES) val = *(const v2f*)(W2 + c * HIDDEN + kk);
        *(v2f*)&W2s[c][kk] = val;
    }
    __syncthreads();

    // neighbor mean: 8 threads per node, 16 h-values per thread (float2 x8)
    {
        int m  = tid >> 3;
        int hb = (tid & 7) * 16;
        v2f s[8];
        #pragma unroll
        for (int k = 0; k < 8; ++k) s[k] = (v2f){0.f, 0.f};
        for (int j = 0; j < FANOUT; ++j) {
            const float* p = h1 + (long)nidx[m * FANOUT + j] * HIDDEN + hb;
            #pragma unroll
            for (int k = 0; k < 8; ++k) {
                v2f v = *(const v2f*)(p + 2 * k);
                s[k].x += v.x; s[k].y += v.y;
            }
        }
        const float inv = 1.0f / (float)FANOUT;
        #pragma unroll
        for (int k = 0; k < 8; ++k) {
            v2f v = { s[k].x * inv, s[k].y * inv };
            *(v2f*)&Agg[m][hb + 2 * k] = v;
        }
    }
    __syncthreads();

    // final GEMM: 3 waves x (16x16 tile), K=128 -> 32 f32 WMMA steps
    if (wave < 3) {                      // wave-uniform: EXEC all-1s in WMMA
        int n0 = wave * 16;
        v8f acc = {};
        #pragma unroll
        for (int ks = 0; ks < 32; ++ks) {
            int k0 = ks * 4 + 2 * hi;
            v2f a = *(const v2f*)&Agg[lm][k0];
            v2f b = *(const v2f*)&W2s[n0 + lm][k0];
            acc = __builtin_amdgcn_wmma_f32_16x16x4_f32(
                false, a, false, b, (short)0, acc, false, false);
        }
        int cc = n0 + lm;
        if (cc < CLASSES) {
            float bias = b2[cc];
            #pragma unroll
            for (int r = 0; r < 8; ++r) {
                int m = r + 8 * hi;
                out[(long)(mbase + m) * CLASSES + cc] = acc[r] + bias;
            }
        }
    }
}

extern "C" void kernel_launch(void* const* d_in, const int* in_sizes, int n_in,
                              void* d_out, int out_size, void* d_ws, size_t ws_size,
                              hipStream_t stream) {
    const float* features = (const float*)d_in[0];
    const float* W1       = (const float*)d_in[1];
    const float* b1       = (const float*)d_in[2];
    const float* W2       = (const float*)d_in[3];
    const float* b2       = (const float*)d_in[4];
    const int*   map1     = (const int*)d_in[5];
    const int*   neigh    = (const int*)d_in[6];
    float* outp = (float*)d_out;
    float* h1   = (float*)d_ws;   // N1*HIDDEN floats = 25.6 MB scratch

    gcn_l1_gather_gemm_relu<<<N1 / 16, 256, 0, stream>>>(features, W1, b1, map1, h1);
    gcn_l2_agg_gemm<<<N2 / 16, 128, 0, stream>>>(h1, W2, b2, neigh, outp);
}